// KVPosAttentionMapping_56727928046101
// MI455X (gfx1250) — compile-verified
//
#include <hip/hip_runtime.h>
#include <hip/hip_bf16.h>

typedef __attribute__((ext_vector_type(2))) float v2f;
typedef __attribute__((ext_vector_type(8))) float v8f;

#define NSEQ 1024
#define DK   64
#define NBH  128   // B*H = 8*16

// ---------------------------------------------------------------------------
// sum(w_pos) / sqrt(n_heads) -> ws[0]
// ---------------------------------------------------------------------------
__global__ void sumw_kernel(const float* __restrict__ w, float* __restrict__ out) {
    __shared__ float s[64];
    s[threadIdx.x] = w[threadIdx.x];
    __syncthreads();
    if (threadIdx.x == 0) {
        float a = 0.f;
#pragma unroll
        for (int i = 0; i < 64; ++i) a += s[i];
        out[0] = a * 0.25f;   // / sqrt(16)
    }
}

// ---------------------------------------------------------------------------
// pos_bias[n*1024+m] = dot(pos_enc[n,m,:], w_pos) + b_pos   (streams 256 MB once)
// ---------------------------------------------------------------------------
__global__ void __launch_bounds__(256)
bias_kernel(const float* __restrict__ pe, const float* __restrict__ w,
            const float* __restrict__ b, float* __restrict__ bias) {
    size_t idx = (size_t)blockIdx.x * 256 + threadIdx.x;      // over 1024*1024
    const float4* p  = (const float4*)(pe + idx * 64);
    const float4* wv = (const float4*)w;                      // uniform -> scalar loads
    float acc = 0.f;
#pragma unroll
    for (int k = 0; k < 16; ++k) {
        float4 x = p[k];
        float4 ww = wv[k];
        acc += x.x * ww.x + x.y * ww.y + x.z * ww.z + x.w * ww.w;
    }
    bias[idx] = acc + b[0];
}

// ---------------------------------------------------------------------------
// Fragment load for C = K*K^T with V_WMMA_F32_16X16X4_F32.
// A (16x4): lane L<16 holds row L, K=kk..kk+1 ; lane L+16 holds row L, K=kk+2..kk+3.
// B (4x16) = K[cols]^T has the mirrored layout -> identical load pattern.
// ---------------------------------------------------------------------------
__device__ __forceinline__ v2f frag_ld(const float* __restrict__ Kb, int base,
                                       int kk, int lane) {
    const float* p = Kb + (size_t)(base + (lane & 15)) * DK + kk + ((lane >> 4) << 1);
    float2 t = *(const float2*)p;     // 8B-aligned (kk even)
    v2f f;
    f.x = t.x;
    f.y = t.y;
    return f;
}

// ---------------------------------------------------------------------------
// attn[bh,n,m] = (K[n]. K[m]) * (sum_w/4) + pos_bias[n,m]
// block = 256 threads (8 waves), computes a 128x128 tile of one bh's 1024x1024.
// wave = 64x32 sub-tile = 4x2 WMMA fragments.
// ---------------------------------------------------------------------------
__global__ void __launch_bounds__(256)
gemm_kernel(const float* __restrict__ keys, const float* __restrict__ ws,
            float* __restrict__ out) {
    const int lane = threadIdx.x & 31;
    const int wave = threadIdx.x >> 5;                        // 0..7
    const int bh   = blockIdx.z;
    const int rowT = blockIdx.y * 128 + (wave >> 2) * 64;     // n base for this wave
    const int colT = blockIdx.x * 128 + (wave & 3) * 32;      // m base for this wave

    const float* Kb    = keys + (size_t)bh * NSEQ * DK;
    const float  scale = ws[0];
    const float* bias  = ws + 16;

    v8f acc[4][2];
#pragma unroll
    for (int i = 0; i < 4; ++i)
#pragma unroll
        for (int j = 0; j < 2; ++j)
            acc[i][j] = (v8f)(0.f);

    for (int kk = 0; kk < DK; kk += 4) {
        v2f a[4], b[2];
#pragma unroll
        for (int i = 0; i < 4; ++i) a[i] = frag_ld(Kb, rowT + i * 16, kk, lane);
#pragma unroll
        for (int j = 0; j < 2; ++j) b[j] = frag_ld(Kb, colT + j * 16, kk, lane);
#pragma unroll
        for (int i = 0; i < 4; ++i)
#pragma unroll
            for (int j = 0; j < 2; ++j)
                acc[i][j] = __builtin_amdgcn_wmma_f32_16x16x4_f32(
                    false, a[i], false, b[j], (short)0, acc[i][j], false, false);
    }

    // C layout: VGPR g -> row (g + 8*(lane>=16)), col (lane&15)
    float* outb = out + (size_t)bh * NSEQ * NSEQ;
    const int rsub = (lane >> 4) << 3;
    const int csub = lane & 15;
#pragma unroll
    for (int i = 0; i < 4; ++i) {
#pragma unroll
        for (int j = 0; j < 2; ++j) {
            const int c = colT + j * 16 + csub;
#pragma unroll
            for (int g = 0; g < 8; ++g) {
                const int r = rowT + i * 16 + rsub + g;
                const size_t rc = (size_t)r * NSEQ + c;
                outb[rc] = acc[i][j][g] * scale + bias[rc];
            }
        }
    }
}

// ---------------------------------------------------------------------------
extern "C" void kernel_launch(void* const* d_in, const int* in_sizes, int n_in,
                              void* d_out, int out_size, void* d_ws, size_t ws_size,
                              hipStream_t stream) {
    (void)in_sizes; (void)n_in; (void)out_size; (void)ws_size;

    const float* queries = (const float*)d_in[0];
    const float* keys    = (const float*)d_in[1];
    const float* values  = (const float*)d_in[2];
    const float* pos_enc = (const float*)d_in[3];
    const float* w_pos   = (const float*)d_in[4];
    const float* b_pos   = (const float*)d_in[5];

    float* out = (float*)d_out;
    float* ws  = (float*)d_ws;   // ws[0]=scale, ws[16..16+1M) = pos_bias (4 MB)

    const size_t attnElems = (size_t)NBH * NSEQ * NSEQ;        // 134,217,728
    const size_t qkvElems  = (size_t)NBH * NSEQ * DK;          //   8,388,608

    // pass-through q, k, v (d2d copies; graph-capture safe)
    hipMemcpyAsync(out + attnElems,                queries, qkvElems * sizeof(float),
                   hipMemcpyDeviceToDevice, stream);
    hipMemcpyAsync(out + attnElems + qkvElems,     keys,    qkvElems * sizeof(float),
                   hipMemcpyDeviceToDevice, stream);
    hipMemcpyAsync(out + attnElems + 2 * qkvElems, values,  qkvElems * sizeof(float),
                   hipMemcpyDeviceToDevice, stream);

    sumw_kernel<<<1, 64, 0, stream>>>(w_pos, ws);
    bias_kernel<<<(NSEQ * NSEQ) / 256, 256, 0, stream>>>(pos_enc, w_pos, b_pos, ws + 16);

    dim3 grid(NSEQ / 128, NSEQ / 128, NBH);
    gemm_kernel<<<grid, 256, 0, stream>>>(keys, ws, out);
}